// Hiruma_71184787964128
// MI455X (gfx1250) — compile-verified
//
#include <hip/hip_runtime.h>
#include <hip/hip_bf16.h>

typedef __attribute__((ext_vector_type(2))) float v2f;
typedef __attribute__((ext_vector_type(8))) float v8f;
typedef __attribute__((ext_vector_type(4))) unsigned int u32x4;
typedef __attribute__((ext_vector_type(8))) int i32x8;
typedef __attribute__((ext_vector_type(4))) int i32x4;

#ifndef HAVE_TDM
#ifdef __has_builtin
#if __has_builtin(__builtin_amdgcn_tensor_load_to_lds)
#define HAVE_TDM 1
#endif
#endif
#endif
#ifndef HAVE_TDM
#define HAVE_TDM 0
#endif

// ---------------------------------------------------------------------------
// TDM: 1-D contiguous copy of nfloats fp32 from global to LDS via
// TENSOR_LOAD_TO_LDS. D# built per CDNA5 ISA ch.8:
//   group0: count=1, lds_addr, global_addr[56:0], type=2
//   group1: data_size=4B, tensor_dim0=tile_dim0=n (n < 2^16), tensor_dim1=1,
//           tile_dim1=1, stride0=n, no padding/iteration/multicast.
// Wave-level op (EXEC ignored): caller must issue from exactly one wave and
// follow with s_wait_tensorcnt 0 + barrier before LDS reads.
// ---------------------------------------------------------------------------
#if HAVE_TDM
__device__ __forceinline__ void tdm_copy_to_lds(float* lds_dst,
                                                const float* src, int nfloats)
{
  unsigned long long ga = (unsigned long long)(uintptr_t)src;
  unsigned int n = (unsigned int)nfloats;

  u32x4 g0;
  g0.x = 1u;                                              // count=1
  g0.y = (unsigned int)(uintptr_t)lds_dst;                // LDS byte address
  g0.z = (unsigned int)(ga & 0xFFFFFFFFu);                // global_addr[31:0]
  g0.w = (unsigned int)((ga >> 32) & 0x01FFFFFFu)         // global_addr[56:32]
         | 0x80000000u;                                   // type=2 ("image")

  i32x8 g1;
  g1[0] = (int)0x00020000u;                 // data_size=2 (4 bytes)
  g1[1] = (int)((n & 0xFFFFu) << 16);       // tensor_dim0[15:0]  @bits[63:48]
  g1[2] = (int)(((n >> 16) & 0xFFFFu)       // tensor_dim0[31:16] @bits[79:64]
                | (1u << 16));              // tensor_dim1=1      @bits[95:80]
  g1[3] = (int)((n & 0xFFFFu) << 16);       // tile_dim0=n        @bits[127:112]
  g1[4] = (int)1;                           // tile_dim1=1, tile_dim2=0
  g1[5] = (int)n;                           // tensor_dim0_stride[31:0]
  g1[6] = 0;                                // stride0 hi, stride1 lo
  g1[7] = 0;                                // stride1 hi

  i32x4 z4 = {0, 0, 0, 0};
#if __clang_major__ >= 23
  i32x8 z8 = {0, 0, 0, 0, 0, 0, 0, 0};
  __builtin_amdgcn_tensor_load_to_lds(g0, g1, z4, z4, z8, 0);
#else
  __builtin_amdgcn_tensor_load_to_lds(g0, g1, z4, z4, 0);
#endif
}
#endif

// ---------------------------------------------------------------------------
// conv1: 5ch (3 img + 2 coord) -> 16ch, 3x3 VALID, ReLU. Direct VALU kernel
// (K=45 too small/odd for WMMA; ~1.5 GF). Coord channels analytic.
// ---------------------------------------------------------------------------
__global__ __launch_bounds__(256) void k_conv1(const float* __restrict__ img,
                                               const float* __restrict__ w1,
                                               const float* __restrict__ b1,
                                               float* __restrict__ out, int B)
{
  constexpr int H = 128, W = 128, OH = 126, OW = 126, CO = 16;
  long idx = (long)blockIdx.x * 256 + threadIdx.x;
  long total = (long)B * CO * OH * OW;
  if (idx >= total) return;
  int ox = (int)(idx % OW); long t = idx / OW;
  int oy = (int)(t % OH); t /= OH;
  int co = (int)(t % CO); int b = (int)(t / CO);

  const float* ib = img + (long)b * 3 * H * W;
  const float* wc = w1 + co * 5 * 9;
  float acc = b1[co];
  #pragma unroll
  for (int ci = 0; ci < 3; ci++)
    #pragma unroll
    for (int dy = 0; dy < 3; dy++)
      #pragma unroll
      for (int dx = 0; dx < 3; dx++)
        acc += ib[(long)ci * H * W + (long)(oy + dy) * W + (ox + dx)] *
               wc[ci * 9 + dy * 3 + dx];
  #pragma unroll
  for (int dy = 0; dy < 3; dy++) {
    float xx = -1.0f + (2.0f / (H - 1)) * (float)(oy + dy);
    #pragma unroll
    for (int dx = 0; dx < 3; dx++) {
      float yy = -1.0f + (2.0f / (W - 1)) * (float)(ox + dx);
      acc += xx * wc[3 * 9 + dy * 3 + dx] + yy * wc[4 * 9 + dy * 3 + dx];
    }
  }
  out[idx] = acc > 0.0f ? acc : 0.0f;
}

// ---------------------------------------------------------------------------
// 3x3 VALID conv + bias + ReLU as implicit GEMM on V_WMMA_F32_16X16X4_F32.
//   M = cout (padded to MT*16), N = 16 output pixels along ox, K = CIN*9.
// Weights staged to LDS by TDM (TENSOR_LOAD_TO_LDS); im2col offsets come
// from a K-entry LDS table so the hot loop is just ds_load + 2 global loads
// + wmma. fp32 A/B lane layout per CDNA5 ISA:
//   VGPR0: K=k0   (lanes 0-15) / K=k0+2 (lanes 16-31)
//   VGPR1: K=k0+1 (lanes 0-15) / K=k0+3 (lanes 16-31)
// C/D: VGPR i -> M = i (lanes 0-15) / 8+i (lanes 16-31), N = lane&15.
// ---------------------------------------------------------------------------
template <int CIN, int COUT, int H, int W>
__global__ __launch_bounds__(256) void k_conv3x3_wmma(
    const float* __restrict__ in, const float* __restrict__ wgt,
    const float* __restrict__ bias, float* __restrict__ out, int B)
{
  constexpr int OH = H - 2, OW = W - 2, K = CIN * 9;
  constexpr int MT = (COUT + 15) / 16, MPAD = MT * 16;
  constexpr int NTX = (OW + 15) / 16;

  __shared__ float wl[MPAD * K];
  __shared__ int offt[K];

  // im2col offset table: off[k] = ci*H*W + dy*W + dx
  for (int i = threadIdx.x; i < K; i += 256) {
    int ci = i / 9, r = i - ci * 9;
    offt[i] = ci * H * W + (r / 3) * W + (r - (r / 3) * 3);
  }
  // zero the cout-padding rows
  for (int i = COUT * K + (int)threadIdx.x; i < MPAD * K; i += 256)
    wl[i] = 0.0f;
#if HAVE_TDM
  if ((threadIdx.x >> 5) == 0) {            // wave 0 issues the DMA
    tdm_copy_to_lds(wl, wgt, COUT * K);
    __builtin_amdgcn_s_wait_tensorcnt(0);
  }
#else
  for (int i = threadIdx.x; i < COUT * K; i += 256) wl[i] = wgt[i];
#endif
  __syncthreads();

  const int wave = threadIdx.x >> 5, lane = threadIdx.x & 31;
  const int lm = lane & 15, lh = lane >> 4;

  long task = (long)blockIdx.x * 8 + wave;
  if (task >= (long)B * OH * NTX * MT) return;   // uniform per wave
  int mt = (int)(task % MT); long t = task / MT;
  int tx = (int)(t % NTX); t /= NTX;
  int oy = (int)(t % OH); int b = (int)(t / OH);

  int px  = tx * 16 + lm;                 // output pixel x (this lane's N col)
  int pxc = px < OW ? px : OW - 1;        // clamp for loads; store is masked
  const float* ibp = in + (long)b * CIN * H * W + (long)oy * W + pxc;

  v8f acc = {0.f, 0.f, 0.f, 0.f, 0.f, 0.f, 0.f, 0.f};
  const int arow = (mt * 16 + lm) * K + 2 * lh;
  const int koff = 2 * lh;

  #pragma unroll 4
  for (int k0 = 0; k0 < K; k0 += 4) {
    v2f a;
    a.x = wl[arow + k0];
    a.y = wl[arow + k0 + 1];
    int o0 = offt[k0 + koff];
    int o1 = offt[k0 + koff + 1];
    v2f bf;
    bf.x = ibp[o0];
    bf.y = ibp[o1];
    acc = __builtin_amdgcn_wmma_f32_16x16x4_f32(
        false, a, false, bf, (short)0, acc, false, false);
  }

  if (px < OW) {
    float* ob = out + (long)b * COUT * OH * OW + (long)oy * OW + px;
    #pragma unroll
    for (int i = 0; i < 8; i++) {
      int co = mt * 16 + lh * 8 + i;
      if (co < COUT) {
        float v = acc[i] + bias[co];
        ob[(long)co * OH * OW] = v > 0.0f ? v : 0.0f;
      }
    }
  }
}

// ---------------------------------------------------------------------------
// 1x1 conv (40->40), no bias/ReLU, as WMMA GEMM: M = cout (pad 48), N = 16
// flat pixels, K = 40. Weights staged via TDM.
// ---------------------------------------------------------------------------
template <int CC>
__global__ __launch_bounds__(256) void k_conv1x1_wmma(
    const float* __restrict__ in, const float* __restrict__ wgt,
    float* __restrict__ out, int B, int HW)
{
  constexpr int K = CC, MT = (CC + 15) / 16, MPAD = MT * 16;
  __shared__ float wl[MPAD * K];
  for (int i = CC * K + (int)threadIdx.x; i < MPAD * K; i += 256) wl[i] = 0.0f;
#if HAVE_TDM
  if ((threadIdx.x >> 5) == 0) {
    tdm_copy_to_lds(wl, wgt, CC * K);
    __builtin_amdgcn_s_wait_tensorcnt(0);
  }
#else
  for (int i = threadIdx.x; i < CC * K; i += 256) wl[i] = wgt[i];
#endif
  __syncthreads();

  const int wave = threadIdx.x >> 5, lane = threadIdx.x & 31;
  const int lm = lane & 15, lh = lane >> 4;
  int NT = (HW + 15) / 16;

  long task = (long)blockIdx.x * 8 + wave;
  if (task >= (long)B * NT * MT) return;
  int mt = (int)(task % MT); long t = task / MT;
  int pt = (int)(t % NT); int b = (int)(t / NT);

  int p  = pt * 16 + lm;
  int pc = p < HW ? p : HW - 1;
  const float* ib = in + (long)b * CC * HW;

  v8f acc = {0.f, 0.f, 0.f, 0.f, 0.f, 0.f, 0.f, 0.f};
  const int arow = (mt * 16 + lm) * K + 2 * lh;

  #pragma unroll 2
  for (int k0 = 0; k0 < K; k0 += 4) {
    if (k0 + 4 < K)
      __builtin_prefetch(&ib[(long)(k0 + 4 + 2 * lh) * HW + pc], 0, 0);
    v2f a;
    a.x = wl[arow + k0];
    a.y = wl[arow + k0 + 1];
    v2f bf;
    bf.x = ib[(long)(k0 + 2 * lh) * HW + pc];
    bf.y = ib[(long)(k0 + 2 * lh + 1) * HW + pc];
    acc = __builtin_amdgcn_wmma_f32_16x16x4_f32(
        false, a, false, bf, (short)0, acc, false, false);
  }

  if (p < HW) {
    float* ob = out + (long)b * CC * HW + p;
    #pragma unroll
    for (int i = 0; i < 8; i++) {
      int co = mt * 16 + lh * 8 + i;
      if (co < CC) ob[(long)co * HW] = acc[i];
    }
  }
}

// ---------------------------------------------------------------------------
// scores[b,q,p] = (1/122) * sum_d query[q,d] * kconv[b, d*4+q, p]
// (scale = sqrt(H*W) = sqrt(14884) = 122 exactly)
// ---------------------------------------------------------------------------
__global__ __launch_bounds__(256) void k_scores(const float* __restrict__ kc,
                                                const float* __restrict__ query,
                                                float* __restrict__ sc,
                                                int B, int HW)
{
  long idx = (long)blockIdx.x * 256 + threadIdx.x;
  long total = (long)B * 4 * HW;
  if (idx >= total) return;
  int p = (int)(idx % HW); long t = idx / HW;
  int q = (int)(t % 4); int b = (int)(t / 4);
  const float* kb = kc + (long)b * 40 * HW + p;
  float acc = 0.0f;
  #pragma unroll
  for (int d = 0; d < 10; d++)
    acc += query[q * 10 + d] * kb[(long)(d * 4 + q) * HW];
  sc[idx] = acc * (1.0f / 122.0f);
}

// ---------------------------------------------------------------------------
// Per (b,q): softmax over HW, write att_map, and reduce pt_out (2) +
// feat_out (10) attention-weighted sums. One 256-thread block per row.
// ---------------------------------------------------------------------------
__global__ __launch_bounds__(256) void k_att(const float* __restrict__ sc,
                                             const float* __restrict__ fc,
                                             float* __restrict__ att,
                                             float* __restrict__ pt_out,
                                             float* __restrict__ feat_out,
                                             int HW, int W)
{
  __shared__ float red[256];
  const int tid = threadIdx.x;
  const int b = blockIdx.x >> 2, q = blockIdx.x & 3;
  const float* s = sc + (long)(b * 4 + q) * HW;

  float m = -3.402823e38f;
  for (int p = tid; p < HW; p += 256) m = fmaxf(m, s[p]);
  red[tid] = m; __syncthreads();
  for (int st = 128; st > 0; st >>= 1) {
    if (tid < st) red[tid] = fmaxf(red[tid], red[tid + st]);
    __syncthreads();
  }
  m = red[0]; __syncthreads();

  float sum = 0.0f;
  for (int p = tid; p < HW; p += 256) sum += __expf(s[p] - m);
  red[tid] = sum; __syncthreads();
  for (int st = 128; st > 0; st >>= 1) {
    if (tid < st) red[tid] += red[tid + st];
    __syncthreads();
  }
  float inv = 1.0f / red[0]; __syncthreads();

  float* ao = att + (long)(b * 4 + q) * HW;
  const float* fb = fc + (long)b * 40 * HW;
  float accs[12];
  #pragma unroll
  for (int i = 0; i < 12; i++) accs[i] = 0.0f;
  const float wx = 1.0f / (float)(W - 1);

  for (int p = tid; p < HW; p += 256) {
    float a = __expf(s[p] - m) * inv;
    ao[p] = a;
    int px = p % W, py = p / W;            // H == W == 122
    accs[0] += a * ((float)px * wx);
    accs[1] += a * ((float)py * wx);
    #pragma unroll
    for (int d = 0; d < 10; d++)
      accs[2 + d] += a * fb[(long)(d * 4 + q) * HW + p];
  }

  for (int r = 0; r < 12; r++) {
    red[tid] = accs[r]; __syncthreads();
    for (int st = 128; st > 0; st >>= 1) {
      if (tid < st) red[tid] += red[tid + st];
      __syncthreads();
    }
    if (tid == 0) {
      if (r == 0)      pt_out[b * 8 + q * 2]     = red[0];
      else if (r == 1) pt_out[b * 8 + q * 2 + 1] = red[0];
      else             feat_out[b * 40 + q * 10 + (r - 2)] = red[0];
    }
    __syncthreads();
  }
}

// ---------------------------------------------------------------------------
extern "C" void kernel_launch(void* const* d_in, const int* in_sizes, int n_in,
                              void* d_out, int out_size, void* d_ws, size_t ws_size,
                              hipStream_t stream)
{
  (void)in_sizes; (void)n_in; (void)out_size; (void)ws_size;

  const float* img   = (const float*)d_in[0];
  const float* w1    = (const float*)d_in[1];
  const float* b1    = (const float*)d_in[2];
  const float* w2    = (const float*)d_in[3];
  const float* b2    = (const float*)d_in[4];
  const float* w3    = (const float*)d_in[5];
  const float* b3    = (const float*)d_in[6];
  const float* k_w   = (const float*)d_in[7];
  const float* f_w   = (const float*)d_in[8];
  const float* query = (const float*)d_in[9];

  const int B = 64;
  const int HW = 122 * 122;                       // 14884

  // Output tuple, concatenated flat: x, pt_out, feat_out, att_map
  float* out   = (float*)d_out;
  float* xo    = out;                             // (B,40,122,122)
  float* pt_o  = out + (long)B * 40 * HW;         // (B,8)
  float* ft_o  = pt_o + (long)B * 8;              // (B,40)
  float* att_o = ft_o + (long)B * 40;             // (B,4,1,HW)

  // Workspace layout (floats)
  float* ws = (float*)d_ws;
  float* c1 = ws;                                 // (B,16,126,126)
  float* c2 = c1 + (long)B * 16 * 126 * 126;      // (B,32,124,124)
  float* kc = c2 + (long)B * 32 * 124 * 124;      // (B,40,HW)
  float* fc = kc + (long)B * 40 * HW;             // (B,40,HW)
  float* sc = fc + (long)B * 40 * HW;             // (B,4,HW)

  {
    long total = (long)B * 16 * 126 * 126;
    k_conv1<<<(int)((total + 255) / 256), 256, 0, stream>>>(img, w1, b1, c1, B);
  }
  {
    long waves = (long)B * 124 * 8 * 2;           // OH*NTX*MT
    k_conv3x3_wmma<16, 32, 126, 126>
        <<<(int)((waves + 7) / 8), 256, 0, stream>>>(c1, w2, b2, c2, B);
  }
  {
    long waves = (long)B * 122 * 8 * 3;
    k_conv3x3_wmma<32, 40, 124, 124>
        <<<(int)((waves + 7) / 8), 256, 0, stream>>>(c2, w3, b3, xo, B);
  }
  {
    int NT = (HW + 15) / 16;
    long waves = (long)B * NT * 3;
    int blocks = (int)((waves + 7) / 8);
    k_conv1x1_wmma<40><<<blocks, 256, 0, stream>>>(xo, k_w, kc, B, HW);
    k_conv1x1_wmma<40><<<blocks, 256, 0, stream>>>(xo, f_w, fc, B, HW);
  }
  {
    long total = (long)B * 4 * HW;
    k_scores<<<(int)((total + 255) / 256), 256, 0, stream>>>(kc, query, sc, B, HW);
  }
  k_att<<<B * 4, 256, 0, stream>>>(sc, fc, att_o, pt_o, ft_o, HW, 122);
}